// TaskSpecificReadout_32615981646210
// MI455X (gfx1250) — compile-verified
//
#include <hip/hip_runtime.h>
#include <hip/hip_bf16.h>

// Problem constants (match reference)
#define BSZ   2048
#define DDIM  1024
#define TNUM  16
#define CDIM  128
#define NTILES (BSZ / 16 + TNUM)   // 128 + 16 = 144 (worst-case padded tiles)
#define KCHUNK 256
#define XSTRIDE (KCHUNK + 4)       // LDS pad: 260 mod 64 = 4 -> conflict-free A reads

typedef __attribute__((ext_vector_type(2))) float v2f;
typedef __attribute__((ext_vector_type(8))) float v8f;

// ---------------------------------------------------------------------------
// Kernel 1: group samples by task into 16-padded segments.
//   order[NTILES*16]  : sample index per tile slot, -1 = padding
//   tile_task[NTILES] : task id per tile, -1 = unused tile
// ---------------------------------------------------------------------------
__global__ void tsr_route_kernel(const int* __restrict__ tasks,
                                 int* __restrict__ order,
                                 int* __restrict__ tile_task) {
    __shared__ int cnt[TNUM];
    __shared__ int base[TNUM];
    __shared__ int cursor[TNUM];

    const int tid = threadIdx.x;
    if (tid < TNUM) { cnt[tid] = 0; cursor[tid] = 0; }
    for (int i = tid; i < NTILES * 16; i += blockDim.x) order[i] = -1;
    __syncthreads();

    for (int n = tid; n < BSZ; n += blockDim.x)
        atomicAdd(&cnt[tasks[n]], 1);
    __syncthreads();

    if (tid == 0) {
        int run = 0;
        for (int t = 0; t < TNUM; ++t) {
            base[t] = run;
            int tiles = (cnt[t] + 15) >> 4;
            int t0 = run >> 4;
            for (int j = 0; j < tiles; ++j) tile_task[t0 + j] = t;
            run += tiles << 4;
        }
        for (int j = run >> 4; j < NTILES; ++j) tile_task[j] = -1;
    }
    __syncthreads();

    for (int n = tid; n < BSZ; n += blockDim.x) {
        int t = tasks[n];
        int pos = base[t] + atomicAdd(&cursor[t], 1);
        order[pos] = n;
    }
}

// ---------------------------------------------------------------------------
// Kernel 2: per-tile GEMM via V_WMMA_F32_16X16X4_F32 (fp32 matrix pipe).
// Block = 256 threads = 8 waves; wave w computes cols [16w, 16w+16).
// ---------------------------------------------------------------------------
__global__ void __launch_bounds__(256)
tsr_gemm_kernel(const float* __restrict__ x,
                const float* __restrict__ weights,
                const float* __restrict__ biases,
                const int* __restrict__ order,
                const int* __restrict__ tile_task,
                float* __restrict__ out) {
    __shared__ float xs[16 * XSTRIDE];   // 16 rows x (256+4) f32 per K-chunk
    __shared__ int   rowidx[16];

    const int tile = blockIdx.x;
    const int t = tile_task[tile];
    if (t < 0) return;                    // uniform across block: safe early-out

    const int tid  = threadIdx.x;
    const int lane = tid & 31;
    const int wave = tid >> 5;

    if (tid < 16) rowidx[tid] = order[tile * 16 + tid];
    __syncthreads();

    // WMMA f32 16x16x4 fragment coordinates (wave32):
    //   A: lane m = lane&15 holds row m; K pair = (lane>>4)*2 .. +1
    //   B: lane n = lane&15 holds col n; same K pair
    const int m    = lane & 15;
    const int n    = lane & 15;
    const int koff = (lane >> 4) << 1;
    const int n0   = wave << 4;           // this wave's first output column

    const float* wbase =
        weights + ((size_t)t * CDIM + (size_t)(n0 + n)) * DDIM + koff;

    v8f acc = {};

    for (int c = 0; c < DDIM / KCHUNK; ++c) {
        // Cooperative gather of the 16 x-rows for this K-chunk into LDS.
        for (int i = tid; i < 16 * (KCHUNK / 4); i += 256) {
            const int r  = i >> 6;              // row 0..15
            const int c4 = i & 63;              // float4 index within chunk
            const int s  = rowidx[r];
            float4 val = make_float4(0.f, 0.f, 0.f, 0.f);
            if (s >= 0)
                val = *(const float4*)(x + (size_t)s * DDIM + c * KCHUNK + c4 * 4);
            *(float4*)(&xs[r * XSTRIDE + c4 * 4]) = val;
        }
        __syncthreads();

        // Prefetch next K-chunk of this wave's weight rows into cache.
        if (c + 1 < DDIM / KCHUNK)
            __builtin_prefetch(wbase + (c + 1) * KCHUNK, 0, 3);

        const float* wp   = wbase + c * KCHUNK;
        const float* xrow = &xs[m * XSTRIDE + koff];

#pragma unroll 8
        for (int k = 0; k < KCHUNK; k += 4) {
            v2f a = *(const v2f*)(xrow + k);    // ds_load_b64, bank-conflict-free
            v2f b = *(const v2f*)(wp + k);      // global_load_b64 (L2-resident)
            acc = __builtin_amdgcn_wmma_f32_16x16x4_f32(
                /*neg_a=*/false, a, /*neg_b=*/false, b,
                /*c_mod=*/(short)0, acc, /*reuse_a=*/false, /*reuse_b=*/false);
        }
        __syncthreads();
    }

    // Epilogue: C/D layout -> VGPR v holds row M = v + (lane>=16 ? 8 : 0).
    const int   ncol = n0 + (lane & 15);
    const float bv   = biases[t * CDIM + ncol];
    union { v8f v; float f[8]; } res;
    res.v = acc;
#pragma unroll
    for (int v = 0; v < 8; ++v) {
        const int M = v + ((lane >> 4) << 3);
        const int s = rowidx[M];
        if (s >= 0)
            out[(size_t)s * CDIM + ncol] = res.f[v] + bv;
    }
}

// ---------------------------------------------------------------------------
extern "C" void kernel_launch(void* const* d_in, const int* in_sizes, int n_in,
                              void* d_out, int out_size, void* d_ws, size_t ws_size,
                              hipStream_t stream) {
    const float* x       = (const float*)d_in[0];
    const int*   tasks   = (const int*)  d_in[1];
    const float* weights = (const float*)d_in[2];
    const float* biases  = (const float*)d_in[3];
    float*       out     = (float*)d_out;

    int* order     = (int*)d_ws;                 // NTILES*16 ints
    int* tile_task = order + NTILES * 16;        // NTILES ints

    tsr_route_kernel<<<1, 256, 0, stream>>>(tasks, order, tile_task);
    tsr_gemm_kernel<<<NTILES, 256, 0, stream>>>(x, weights, biases,
                                                order, tile_task, out);
}